// Middle_Moudle_v3_ablation_meta_58815282152234
// MI455X (gfx1250) — compile-verified
//
#include <hip/hip_runtime.h>

#define CH   640
#define WIN  19
#define NWID 5
#define P    25           // NW*NW
#define BS   100          // (N_WAY*BATCH) * (N_WAY*K_SHOT)
#define META_ELEMS (BS * 2 * CH * P)   // 3,200,000 floats
#define EPSF 1e-8f

typedef float v2f __attribute__((ext_vector_type(2)));
typedef float v8f __attribute__((ext_vector_type(8)));

// ---------------------------------------------------------------------------
// Kernel 1: adaptive avg pool 19x19 -> 5x5 for support & query, writing the
// concatenated meta_input tensor (B,S,2C,5,5) directly into d_out.
// One wave32 per image; 8 images per 256-thread block.
// ---------------------------------------------------------------------------
__global__ void pool_kernel(const float* __restrict__ sup,
                            const float* __restrict__ qry,
                            float* __restrict__ meta) {
    __shared__ float tile[8][WIN * WIN];
    const int lane = threadIdx.x & 31;
    const int wv   = threadIdx.x >> 5;
    const int img  = blockIdx.x * 8 + wv;          // 0 .. 2*BS*CH-1
    const int which = img / (BS * CH);             // 0 = support, 1 = query
    const int rem   = img - which * (BS * CH);
    const int bs    = rem / CH;
    const int c     = rem - bs * CH;

    const float* src = (which == 0 ? sup : qry) + (size_t)(bs * CH + c) * (WIN * WIN);
    for (int t = lane; t < WIN * WIN; t += 32) tile[wv][t] = src[t];
    __syncthreads();

    if (lane < P) {
        // torch adaptive pool bin boundaries for 19 -> 5
        const int starts[NWID] = {0, 3, 7, 11, 15};
        const int ends[NWID]   = {4, 8, 12, 16, 19};
        const int oi = lane / NWID, oj = lane - oi * NWID;
        const int si = starts[oi], ei = ends[oi];
        const int sj = starts[oj], ej = ends[oj];
        float acc = 0.f;
        for (int i = si; i < ei; ++i)
            for (int j = sj; j < ej; ++j)
                acc += tile[wv][i * WIN + j];
        acc /= (float)((ei - si) * (ej - sj));
        meta[((size_t)bs * (2 * CH) + (size_t)which * CH + c) * P + lane] = acc;
    }
}

// ---------------------------------------------------------------------------
// Kernel 2: per (b,s): dots = S^T(25x640) x Q(640x25) via V_WMMA_F32_16X16X4,
// f32 norms, clamped cosine similarity, max over query positions.
// 128 threads = 4 waves; each wave owns one 16x16 tile of the padded 32x32.
// Pad rows/cols handled by clamped index + 0/1 float mask (no exec divergence
// in the K loop -> plain global_load_b32 + v_wmma per step).
// ---------------------------------------------------------------------------
__global__ void cossim_kernel(const float* __restrict__ meta,
                              float* __restrict__ out) {
    __shared__ float dots[32][33];
    __shared__ float nrm[64];                      // [0..24]=|s|, [32..56]=|q|

    const int bs   = blockIdx.x;
    const int tid  = threadIdx.x;
    const int lane = tid & 31;
    const int wv   = tid >> 5;
    const int hi   = (lane >> 4) & 1;              // half-wave select
    const int lp   = lane & 15;
    const int m0   = (wv >> 1) << 4;               // tile row origin
    const int n0   = (wv & 1) << 4;                // tile col origin

    const float* S = meta + (size_t)bs * (2 * CH * P);        // channels [0,C)
    const float* Q = S + (size_t)CH * P;                      // channels [C,2C)
    const int m = m0 + lp;                                     // support position
    const int n = n0 + lp;                                     // query position

    // Hoisted pad handling: clamp index, multiply loaded value by 0/1 mask.
    const int   mc = (m < P) ? m : (P - 1);
    const int   nc = (n < P) ? n : (P - 1);
    const float mmask = (m < P) ? 1.f : 0.f;
    const float nmask = (n < P) ? 1.f : 0.f;
    const float* Sm = S + mc;   // A column base: element k is Sm[k*P]
    const float* Qn = Q + nc;   // B column base: element k is Qn[k*P]

    v8f acc = {};
#pragma unroll 4
    for (int k0 = 0; k0 < CH; k0 += 4) {
        const int ka = k0 + 2 * hi;
        v2f a, b;
        // A 16x4 f32 layout: lanes0-15 M=lane K={k0,k0+1}; lanes16-31 K={k0+2,k0+3}
        a.x = Sm[ka * P]       * mmask;
        a.y = Sm[(ka + 1) * P] * mmask;
        // B 4x16 f32 layout mirrored
        b.x = Qn[ka * P]       * nmask;
        b.y = Qn[(ka + 1) * P] * nmask;
        acc = __builtin_amdgcn_wmma_f32_16x16x4_f32(
            /*neg_a=*/false, a, /*neg_b=*/false, b,
            /*c_mod=*/(short)0, acc, /*reuse_a=*/false, /*reuse_b=*/false);
    }

    // C/D layout: VGPR r, lanes0-15 -> M=m0+r, lanes16-31 -> M=m0+8+r; N=n0+lp
#pragma unroll
    for (int r = 0; r < 8; ++r)
        dots[m0 + r + 8 * hi][n0 + lp] = acc[r];

    // Norms in f32 (store sqrt): 25 support + 25 query positions.
    if (tid < P) {
        float s2 = 0.f;
        for (int c = 0; c < CH; ++c) { float v = S[c * P + tid]; s2 += v * v; }
        nrm[tid] = sqrtf(s2);
    } else if (tid >= 32 && tid < 32 + P) {
        const int p = tid - 32;
        float s2 = 0.f;
        for (int c = 0; c < CH; ++c) { float v = Q[c * P + p]; s2 += v * v; }
        nrm[32 + p] = sqrtf(s2);
    }
    __syncthreads();

    if (tid < P) {
        const float sn = nrm[tid];
        float best = -3.402823466e+38f;
        for (int q = 0; q < P; ++q) {
            const float denom = fmaxf(sn * nrm[32 + q], EPSF);
            best = fmaxf(best, dots[tid][q] / denom);
        }
        out[(size_t)bs * P + tid] = best;
    }
}

// ---------------------------------------------------------------------------
extern "C" void kernel_launch(void* const* d_in, const int* in_sizes, int n_in,
                              void* d_out, int out_size, void* d_ws, size_t ws_size,
                              hipStream_t stream) {
    const float* sup = (const float*)d_in[0];
    const float* qry = (const float*)d_in[1];
    float* meta = (float*)d_out;                 // first 3,200,000 floats
    float* out  = meta + META_ELEMS;             // next 2,500 floats

    const int n_images = 2 * BS * CH;            // 128,000
    pool_kernel<<<n_images / 8, 256, 0, stream>>>(sup, qry, meta);
    cossim_kernel<<<BS, 128, 0, stream>>>(meta, out);
}